// RejectionSampler_1322849927624
// MI455X (gfx1250) — compile-verified
//
#include <hip/hip_runtime.h>

#define BB 64
#define SS 8
#define VV 32000
#define NT 512                 // 16 waves of 32
#define NV4 (VV / 4)           // 8000 float4 chunks per row
#define SMEM_BYTES (VV * 4 + 1024)

#ifndef __has_builtin
#define __has_builtin(x) 0
#endif

typedef int v4i __attribute__((ext_vector_type(4)));
typedef __attribute__((address_space(1))) v4i* gptr_v4i;   // global (AS1)
typedef __attribute__((address_space(3))) v4i* lptr_v4i;   // LDS (AS3)
typedef __attribute__((address_space(3))) void lvoid;

// 16B async copy global -> LDS (ASYNCcnt-tracked DMA, bypasses VGPRs)
__device__ __forceinline__ void async_copy16(const void* gsrc, void* ldst) {
#if __has_builtin(__builtin_amdgcn_global_load_async_to_lds_b128)
  __builtin_amdgcn_global_load_async_to_lds_b128(
      (gptr_v4i)const_cast<void*>(gsrc),   // addrspacecast generic -> AS1
      (lptr_v4i)ldst,                      // addrspacecast generic -> AS3
      0, 0);                               // imm offset, imm cpol
#else
  unsigned loff = (unsigned)(unsigned long long)(lvoid*)ldst;
  unsigned long long ga = (unsigned long long)gsrc;
  asm volatile("global_load_async_to_lds_b128 %0, %1, off"
               :: "v"(loff), "v"(ga) : "memory");
#endif
}

__device__ __forceinline__ void wait_async0() {
#if __has_builtin(__builtin_amdgcn_s_wait_asynccnt)
  __builtin_amdgcn_s_wait_asynccnt(0);
#else
  asm volatile("s_wait_asynccnt 0" ::: "memory");
#endif
}

__global__ void __launch_bounds__(NT)
rejection_sampler_kernel(const float* __restrict__ tgt,   // (B, S+1, V)
                         const float* __restrict__ drf,   // (B, S, V)
                         const int*   __restrict__ ids,   // (B, S)
                         const float* __restrict__ ru,    // (B, S)
                         const float* __restrict__ su,    // (B,)
                         int* __restrict__ out)           // (B, S+1)
{
  extern __shared__ unsigned char smem[];
  float* pbuf  = (float*)smem;               // VV floats (row -> residual p, in place)
  float* wsum  = (float*)(smem + VV * 4);    // 16: per-wave partials
  float* woff  = wsum + 16;                  // 16: per-wave scan offsets
  float* fmisc = woff + 16;                  // [0]=total, [1]=running prefix
  int*   iscr  = (int*)(fmisc + 4);          // 16: count partials

  const int b    = blockIdx.x;
  const int tid  = threadIdx.x;
  const int lane = tid & 31;                 // wave32
  const int w    = tid >> 5;                 // 16 waves

  // ---- 1. first rejected position (uniform across all threads; hits cache) ----
  int rej = SS;
  for (int t = 0; t < SS; ++t) {
    int   id = ids[b * SS + t];
    float tt = tgt[((size_t)b * (SS + 1) + t) * VV + id];
    float dd = drf[((size_t)b * SS + t) * VV + id];
    if (tt / dd < ru[b * SS + t]) { rej = t; break; }
  }

  const float4* rowT = (const float4*)(tgt + ((size_t)b * (SS + 1) + rej) * VV);
  const float4* rowD = (rej < SS)
                     ? (const float4*)(drf + ((size_t)b * SS + rej) * VV)
                     : nullptr;

  // ---- 2. async-stage the selected target row into LDS (128 KB) ----
  for (int i4 = tid; i4 < NV4; i4 += NT)
    async_copy16(rowT + i4, (float4*)pbuf + i4);
  wait_async0();
  __syncthreads();

  // ---- 3. pass A: p = max(t - d, 0) in place (bonus row: p = t); total = sum(p) ----
  float lsum = 0.f;
  if (rej < SS) {
    for (int i4 = tid; i4 < NV4; i4 += NT) {
      float4 t4 = ((float4*)pbuf)[i4];
      float4 d4 = rowD[i4];
      float4 p;
      p.x = fmaxf(t4.x - d4.x, 0.f);
      p.y = fmaxf(t4.y - d4.y, 0.f);
      p.z = fmaxf(t4.z - d4.z, 0.f);
      p.w = fmaxf(t4.w - d4.w, 0.f);
      ((float4*)pbuf)[i4] = p;
      lsum += (p.x + p.y) + (p.z + p.w);
    }
  } else {
    for (int i4 = tid; i4 < NV4; i4 += NT) {
      float4 t4 = ((float4*)pbuf)[i4];
      lsum += (t4.x + t4.y) + (t4.z + t4.w);
    }
  }
  for (int o = 16; o; o >>= 1) lsum += __shfl_down(lsum, o, 32);
  if (lane == 0) wsum[w] = lsum;
  __syncthreads();
  if (w == 0) {
    float v = (lane < 16) ? wsum[lane] : 0.f;
    for (int o = 16; o; o >>= 1) v += __shfl_down(v, o, 32);
    if (lane == 0) { fmisc[0] = v; fmisc[1] = 0.f; }
  }
  __syncthreads();
  const float T = su[b] * fmisc[0];   // normalization cancels algebraically

  // ---- 4. pass B: streaming block scan over LDS p; count cdf[v] < T ----
  int lcnt = 0;
  const int NTILES = (NV4 + NT - 1) / NT;   // 16
  for (int tile = 0; tile < NTILES; ++tile) {
    int  i4    = tile * NT + tid;
    bool valid = i4 < NV4;
    float4 p4  = valid ? ((float4*)pbuf)[i4] : make_float4(0.f, 0.f, 0.f, 0.f);
    float ts   = (p4.x + p4.y) + (p4.z + p4.w);
    float incl = ts;
    for (int o = 1; o < 32; o <<= 1) {      // wave32 inclusive scan
      float y = __shfl_up(incl, o, 32);
      if (lane >= o) incl += y;
    }
    if (lane == 31) wsum[w] = incl;
    __syncthreads();
    if (tid == 0) {                         // serial scan of 16 wave sums
      float run = fmisc[1];
      for (int k = 0; k < 16; ++k) { woff[k] = run; run += wsum[k]; }
      fmisc[1] = run;
    }
    __syncthreads();
    if (valid) {
      float c = woff[w] + (incl - ts);      // exclusive prefix for this thread
      c += p4.x; lcnt += (c < T);
      c += p4.y; lcnt += (c < T);
      c += p4.z; lcnt += (c < T);
      c += p4.w; lcnt += (c < T);
    }
  }
  for (int o = 16; o; o >>= 1) lcnt += __shfl_down(lcnt, o, 32);
  if (lane == 0) iscr[w] = lcnt;
  __syncthreads();

  // ---- 5. emit outputs ----
  if (tid == 0) {
    int c = 0;
    for (int k = 0; k < 16; ++k) c += iscr[k];
    int sampled = (c < VV) ? c : (VV - 1);
    for (int t = 0; t <= SS; ++t) {
      int v = (t < rej) ? ids[b * SS + t] : ((t == rej) ? sampled : -1);
      out[b * (SS + 1) + t] = v;
    }
  }
}

extern "C" void kernel_launch(void* const* d_in, const int* in_sizes, int n_in,
                              void* d_out, int out_size, void* d_ws, size_t ws_size,
                              hipStream_t stream) {
  (void)in_sizes; (void)n_in; (void)out_size; (void)d_ws; (void)ws_size;
  const float* tgt = (const float*)d_in[0];
  const float* drf = (const float*)d_in[1];
  const int*   ids = (const int*)d_in[2];
  const float* ru  = (const float*)d_in[3];
  const float* su  = (const float*)d_in[4];
  int* out = (int*)d_out;
  rejection_sampler_kernel<<<BB, NT, SMEM_BYTES, stream>>>(tgt, drf, ids, ru, su, out);
}